// GroupedmHC_82248623718845
// MI455X (gfx1250) — compile-verified
//
#include <hip/hip_runtime.h>

// GroupedmHC on MI455X (gfx1250).
// Roofline: 3 x 128MiB HBM traffic @ 23.3 TB/s => ~17us floor. Projections on
// V_WMMA_F32_16X16X4_F32 (K=4 == group size g); Sinkhorn (exp + 5 row/col
// normalizations of a 4x4) one (token,group) pair per lane on VALU.
// x tile staged to LDS with GLOBAL_LOAD_ASYNC_TO_LDS_B128 (ASYNCcnt path).

#define D_DIM   4096
#define G_NUM   1024
#define TOK_T   16   // tokens per block tile
#define GRP_T   16   // groups per block tile

// LDS arena layout (floats), dynamic shared => region bases are exact offsets.
#define X_OFF   0                        // raw x tile      [tok][64]
#define XN_OFF  1024                     // rms-normed tile [tok][64]
#define RES_OFF 2048                     // res proj  [(tok*16+gl)*17 + n], stride 17 (odd => bank-clean)
#define PP_OFF  (2048 + 16*16*17)        // pre/post proj, same layout, 16 wide
#define SMEM_FLOATS (PP_OFF + 16*16*17)
#define SMEM_BYTES  (SMEM_FLOATS * 4)    // 43008 B

typedef __attribute__((ext_vector_type(2))) float v2f;
typedef __attribute__((ext_vector_type(8))) float v8f;

__global__ __launch_bounds__(256) void mhc_fused_kernel(
    const float* __restrict__ x,        const float* __restrict__ f_out,
    const float* __restrict__ w_rms,
    const float* __restrict__ phi_pre,  const float* __restrict__ phi_post,
    const float* __restrict__ phi_res,
    const float* __restrict__ a_pre,    const float* __restrict__ a_post,
    const float* __restrict__ a_res,
    const float* __restrict__ b_pre,    const float* __restrict__ b_post,
    const float* __restrict__ b_res,
    float* __restrict__ out)
{
    extern __shared__ __align__(16) float smem[];

    const int t       = threadIdx.x;
    const unsigned tokBase = blockIdx.x * TOK_T;   // flat token over B*S
    const unsigned gBase   = blockIdx.y * GRP_T;

    // ---- Phase 0: async-copy x tile (16 tokens x 64 floats) straight to LDS ----
    {
        const unsigned tr = t >> 4, c = t & 15;    // 16 lanes/token row, 256B/row
        const float* gp = x + (size_t)((tokBase + tr) * D_DIM + gBase * 4 + c * 4);
        const unsigned loff = (X_OFF + tr * 64 + c * 4) * 4;
        asm volatile("global_load_async_to_lds_b128 %0, %1, off"
                     :: "v"(loff), "v"(gp) : "memory");
        asm volatile("s_wait_asynccnt 0x0" ::: "memory");
    }
    __syncthreads();

    // ---- Phase 1: per-(token,group) RMSNorm (mean(x^2)+eps), one pair/thread ----
    {
        const int tok = t >> 4, gl = t & 15;
        const float4 xv = *(const float4*)&smem[X_OFF + tok * 64 + gl * 4];
        const float ms  = (xv.x*xv.x + xv.y*xv.y + xv.z*xv.z + xv.w*xv.w) * 0.25f
                        + 1e-5f;
        const float inv = rsqrtf(ms);
        float4 xn;
        xn.x = xv.x * w_rms[0] * inv;
        xn.y = xv.y * w_rms[1] * inv;
        xn.z = xv.z * w_rms[2] * inv;
        xn.w = xv.w * w_rms[3] * inv;
        *(float4*)&smem[XN_OFF + tok * 64 + gl * 4] = xn;
    }
    __syncthreads();

    // ---- Phase 2: WMMA projections. Each of 8 waves owns 2 groups. ----
    // A (16x4 f32): v0 = K0 (lanes 0-15) / K2 (lanes 16-31); v1 = K1 / K3.
    // B (4x16 f32): v0 = K0 / K2 row, v1 = K1 / K3 row (C/D-style row striping).
    // C (16x16 f32): c[v], lane l -> (M = v + 8*(l>>4), N = l&15).
    {
        const unsigned wave = t >> 5, lane = t & 31;
        const unsigned n = lane & 15, hi = lane >> 4;
        const unsigned nm = n & 3;                    // clamped, always in-bounds
        // Arithmetic 0/1 masks: all four phi loads are consumed unconditionally,
        // so the backend cannot sink them into saveexec branches.
        const float fpre  = (n < 4) ? 1.0f : 0.0f;
        const float fpost = (n >= 4 && n < 8) ? 1.0f : 0.0f;
#pragma unroll
        for (unsigned u = 0; u < 2; ++u) {
            const unsigned gl   = wave * 2 + u;
            const unsigned goff = gBase + gl;

            v2f a = *(const v2f*)&smem[XN_OFF + n * 64 + gl * 4 + hi * 2];

            v2f br;                                   // phi_res: [G][4][16]
            br.x = phi_res[goff * 64u + (2u * hi + 0u) * 16u + n];
            br.y = phi_res[goff * 64u + (2u * hi + 1u) * 16u + n];

            const float pre0  = phi_pre [goff * 16u + (2u * hi + 0u) * 4u + nm];
            const float pre1  = phi_pre [goff * 16u + (2u * hi + 1u) * 4u + nm];
            const float post0 = phi_post[goff * 16u + (2u * hi + 0u) * 4u + nm];
            const float post1 = phi_post[goff * 16u + (2u * hi + 1u) * 4u + nm];
            v2f bp;
            bp.x = fpre * pre0 + fpost * post0;
            bp.y = fpre * pre1 + fpost * post1;

            v8f cr = {0.f,0.f,0.f,0.f,0.f,0.f,0.f,0.f};
            v8f cp = {0.f,0.f,0.f,0.f,0.f,0.f,0.f,0.f};
            cr = __builtin_amdgcn_wmma_f32_16x16x4_f32(false, a, false, br,
                                                       (short)0, cr, false, false);
            cp = __builtin_amdgcn_wmma_f32_16x16x4_f32(false, a, false, bp,
                                                       (short)0, cp, false, false);

            // Scatter C: per-lane base + constant offsets (v*272 floats),
            // unconditional stores (no EXEC juggling).
            float* rb = &smem[RES_OFF + ((8u * hi) * 16u + gl) * 17u + n];
            float* pb = &smem[PP_OFF  + ((8u * hi) * 16u + gl) * 17u + n];
#pragma unroll
            for (int v = 0; v < 8; ++v) {
                rb[v * 272] = cr[v];                  // 272 = 16 rows * stride 17
                pb[v * 272] = cp[v];
            }
        }
    }
    __syncthreads();

    // ---- Phase 3: gates + Sinkhorn + residual matvec, one (token,group)/thread ----
    {
        const unsigned tok  = t >> 4, gl = t & 15;
        const unsigned goff = gBase + gl;
        const float* rrow = &smem[RES_OFF + (tok * 16u + gl) * 17u];
        const float* prow = &smem[PP_OFF  + (tok * 16u + gl) * 17u];

        float M[16];
#pragma unroll
        for (int i = 0; i < 16; ++i) {
            const float h = a_res[goff * 16u + i] * rrow[i] + b_res[goff * 16u + i];
            M[i] = __expf(h);
        }
#pragma unroll
        for (int it = 0; it < 5; ++it) {
#pragma unroll
            for (int i = 0; i < 4; ++i) {             // row normalize
                const float s = M[i*4+0] + M[i*4+1] + M[i*4+2] + M[i*4+3] + 1e-9f;
                const float r = 1.0f / s;
                M[i*4+0] *= r; M[i*4+1] *= r; M[i*4+2] *= r; M[i*4+3] *= r;
            }
#pragma unroll
            for (int j = 0; j < 4; ++j) {             // col normalize
                const float s = M[j] + M[4+j] + M[8+j] + M[12+j] + 1e-9f;
                const float r = 1.0f / s;
                M[j] *= r; M[4+j] *= r; M[8+j] *= r; M[12+j] *= r;
            }
        }

        const float4 xv = *(const float4*)&smem[X_OFF + tok * 64u + gl * 4u];
        const float xraw[4] = {xv.x, xv.y, xv.z, xv.w};
        float gated[4], hpost[4];
#pragma unroll
        for (int k = 0; k < 4; ++k) {
            const float hpre_t  = a_pre [goff*4u+k] * prow[k]     + b_pre [goff*4u+k];
            const float hpost_t = a_post[goff*4u+k] * prow[4 + k] + b_post[goff*4u+k];
            const float hpre = 1.0f / (1.0f + __expf(-hpre_t));
            gated[k] = hpre * xraw[k];
            hpost[k] = 2.0f / (1.0f + __expf(-hpost_t));
        }

        const float4 fg = *(const float4*)(f_out
                            + (size_t)((tokBase + tok) * D_DIM + goff * 4u));
        const float fga[4] = {fg.x, fg.y, fg.z, fg.w};
        float o[4];
#pragma unroll
        for (int i = 0; i < 4; ++i) {
            const float r = M[i*4+0]*gated[0] + M[i*4+1]*gated[1]
                          + M[i*4+2]*gated[2] + M[i*4+3]*gated[3];
            o[i] = r + hpost[i] * fga[i];
        }
        *(float4*)(out + (size_t)((tokBase + tok) * D_DIM + goff * 4u))
            = make_float4(o[0], o[1], o[2], o[3]);
    }
}

extern "C" void kernel_launch(void* const* d_in, const int* in_sizes, int n_in,
                              void* d_out, int out_size, void* d_ws, size_t ws_size,
                              hipStream_t stream) {
    (void)in_sizes; (void)n_in; (void)d_ws; (void)ws_size; (void)out_size;
    const float* x        = (const float*)d_in[0];
    const float* f_out    = (const float*)d_in[1];
    const float* w_rms    = (const float*)d_in[2];
    const float* phi_pre  = (const float*)d_in[3];
    const float* phi_post = (const float*)d_in[4];
    const float* phi_res  = (const float*)d_in[5];
    const float* a_pre    = (const float*)d_in[6];
    const float* a_post   = (const float*)d_in[7];
    const float* a_res    = (const float*)d_in[8];
    const float* b_pre    = (const float*)d_in[9];
    const float* b_post   = (const float*)d_in[10];
    const float* b_res    = (const float*)d_in[11];
    float* out = (float*)d_out;

    const int tokens = 4 * 2048;                 // B*S
    dim3 grid(tokens / TOK_T, G_NUM / GRP_T);    // (512, 64)
    dim3 block(256);
    mhc_fused_kernel<<<grid, block, SMEM_BYTES, stream>>>(
        x, f_out, w_rms, phi_pre, phi_post, phi_res,
        a_pre, a_post, a_res, b_pre, b_post, b_res, out);
}